// GCNnet_collective_35527969473093
// MI455X (gfx1250) — compile-verified
//
#include <hip/hip_runtime.h>

typedef __attribute__((ext_vector_type(2))) float v2f;
typedef __attribute__((ext_vector_type(4))) float v4f;
typedef __attribute__((ext_vector_type(8))) float v8f;

#define NG   4
#define NBOX 6
#define NFG  8
#define NFR  256
#define NACT 5
#define EPB  32   // elements per block in main kernel
#define ROWS 8    // padded rows per element (6 real + 2 pad)

// workspace float layout (precompute -> main)
#define WS_M 0     // NG*64 : M_g[i*8+j] = sum_r Wtheta[g,r,i]*Wphi[g,r,j]
#define WS_U 256   // NG*8  : u_g = Wtheta^T b_phi
#define WS_V 288   // NG*8  : v_g = Wphi^T b_theta
#define WS_C 320   // NG    : c_g = b_theta . b_phi

// -------------------------------------------------------------------------
// Kernel 1: collapse the NFR=256 contraction once, on the matrix cores.
// M_g = Wtheta_g^T (8x256) x Wphi_g (256x8), via 64 chained f32 WMMAs.
// Branchless operand construction: pad lanes (i>=8) load a clamped valid
// address and multiply by 0 -> no exec-mask save/restore in the hot loop.
// -------------------------------------------------------------------------
__global__ void gcn_precompute_wmma(const float* __restrict__ Wt,
                                    const float* __restrict__ bth,
                                    const float* __restrict__ Wp,
                                    const float* __restrict__ bph,
                                    float* __restrict__ ws) {
  const int g = threadIdx.x >> 5;   // one wave per relation graph
  const int l = threadIdx.x & 31;
  if (g >= NG) return;
  const float* wt = Wt + g * (NFR * NFG);
  const float* wp = Wp + g * (NFR * NFG);

  const int   i   = l & 15;             // A row / B col (valid < 8, rest pad)
  const int   hi  = l >> 4;             // half-wave selects K {0,1} vs {2,3}
  const float am  = (i < NFG) ? 1.f : 0.f;
  const int   ic  = i & (NFG - 1);

  const float* wtp = wt + 2 * hi * NFG + ic;
  const float* wpp = wp + 2 * hi * NFG + ic;

  v8f acc = {0.f, 0.f, 0.f, 0.f, 0.f, 0.f, 0.f, 0.f};
  for (int s = 0; s < NFR / 4; ++s) {
    v2f a, b;
    // A (16x4): vgpr0 = K {0|2}, vgpr1 = K {1|3};  A[i,r] = Wt[r,i]
    a.x = am * wtp[0];
    a.y = am * wtp[NFG];
    // B (4x16): vgpr0 = rows {0|2}, vgpr1 = rows {1|3}; B[r,j] = Wp[r,j]
    b.x = am * wpp[0];
    b.y = am * wpp[NFG];
    wtp += 4 * NFG;
    wpp += 4 * NFG;
    // (neg_a, A, neg_b, B, c_mod, C, reuse_a, reuse_b)
    acc = __builtin_amdgcn_wmma_f32_16x16x4_f32(false, a, false, b,
                                                (short)0, acc, false, false);
  }

  // D layout: vgpr v, lanes 0..15 -> row v, col = lane. Keep 8x8 block.
  if (l < NFG) {
    float* Mg = ws + WS_M + g * 64;
#pragma unroll
    for (int v = 0; v < NFG; ++v) Mg[v * NFG + l] = acc[v];
  }

  // bias cross terms (zero in practice, cheap to keep general)
  if (l < 8) {
    float u = 0.f;
    for (int r = 0; r < NFR; ++r) u += wt[r * NFG + l] * bph[g * NFR + r];
    ws[WS_U + g * 8 + l] = u;
  } else if (l < 16) {
    const int j = l - 8;
    float v = 0.f;
    for (int r = 0; r < NFR; ++r) v += wp[r * NFG + j] * bth[g * NFR + r];
    ws[WS_V + g * 8 + j] = v;
  } else if (l == 16) {
    float c = 0.f;
    for (int r = 0; r < NFR; ++r) c += bth[g * NFR + r] * bph[g * NFR + r];
    ws[WS_C + g] = c;
  }
}

// -------------------------------------------------------------------------
// Kernel 2: batch pass. 8 lanes per element; feat tile staged through LDS
// once, then the whole per-graph pipeline runs out of registers. Weights
// read as 16B broadcast ds_load_b128; feat rows stride-10 -> b64 loads with
// the four element-groups of a wave on banks {0,16,32,48} (conflict-free).
// -------------------------------------------------------------------------
#define LDS_WEXT 0            // 64   (16B aligned)
#define LDS_BEXT 64           // 8
#define LDS_M    72           // 256  (288B -> 16B aligned)
#define LDS_U    328          // 32
#define LDS_V    360          // 32
#define LDS_C    392          // 4 (+4 pad)
#define LDS_GCN  400          // 256  (1600B -> 16B aligned)
#define LDS_ACT  656          // 40   (2624B -> 16B aligned)
#define LDS_BACT 696          // 5 (+3 pad)
#define FEAT_BASE 704
#define FEAT_STRIDE 10        // even (b64 ok), e-stride 80 -> bank-conflict free
#define POS_BASE (FEAT_BASE + EPB * ROWS * FEAT_STRIDE)   // 3264
#define LDS_TOT  (POS_BASE + EPB * ROWS * 2)              // 3776 floats

__launch_bounds__(256)
__global__ void gcn_main_kernel(const float* __restrict__ bf,
                                const float* __restrict__ bp,
                                const float* __restrict__ Wext,
                                const float* __restrict__ bext,
                                const float* __restrict__ Wgcn,
                                const float* __restrict__ Wact,
                                const float* __restrict__ bact,
                                const float* __restrict__ ws,
                                float* __restrict__ out, int Btot) {
  __shared__ float lds[LDS_TOT];
  const int tid = threadIdx.x;

  // cooperative weight staging (256 threads)
  lds[LDS_M + tid]   = ws[WS_M + tid];     // 256 floats
  lds[LDS_GCN + tid] = Wgcn[tid];          // 256 floats
  if (tid < 64)   lds[LDS_WEXT + tid] = Wext[tid];
  if (tid < 8)    lds[LDS_BEXT + tid] = bext[tid];
  if (tid < 32) { lds[LDS_U + tid] = ws[WS_U + tid];
                  lds[LDS_V + tid] = ws[WS_V + tid]; }
  if (tid < NG)   lds[LDS_C + tid] = ws[WS_C + tid];
  if (tid < 40)   lds[LDS_ACT + tid] = Wact[tid];
  if (tid < NACT) lds[LDS_BACT + tid] = bact[tid];

  const int e_local = tid >> 3;          // 0..31
  const int n       = tid & 7;           // padded row id
  const long e      = (long)blockIdx.x * EPB + e_local;
  const bool rowvalid = (n < NBOX) && (e < (long)Btot);

  v4f x0 = {0.f, 0.f, 0.f, 0.f}, x1 = {0.f, 0.f, 0.f, 0.f};
  float p0 = 0.f, p1 = 0.f;
  if (rowvalid) {
    const v4f* src = (const v4f*)(bf + (e * NBOX + n) * NFG);
    x0 = src[0];
    x1 = src[1];
    const v2f* ps = (const v2f*)(bp + (e * NBOX + n) * 2);
    v2f pp = ps[0];
    p0 = pp.x;
    p1 = pp.y;
  }

  // feat = relu(x W_ext^T + b_ext)
  const v4f* WEXT4 = (const v4f*)&lds[LDS_WEXT];
  float feat[NFG];
#pragma unroll
  for (int o = 0; o < NFG; ++o) {
    v4f w0 = WEXT4[2 * o], w1 = WEXT4[2 * o + 1];
    v4f pr = x0 * w0 + x1 * w1;
    float s = lds[LDS_BEXT + o] + pr[0] + pr[1] + pr[2] + pr[3];
    feat[o] = rowvalid ? fmaxf(s, 0.f) : 0.f;
  }

  // stage own row (feat + position) to the element tile
  {
    v2f* fst = (v2f*)&lds[FEAT_BASE + (e_local * ROWS + n) * FEAT_STRIDE];
    fst[0] = (v2f){feat[0], feat[1]};
    fst[1] = (v2f){feat[2], feat[3]};
    fst[2] = (v2f){feat[4], feat[5]};
    fst[3] = (v2f){feat[6], feat[7]};
    v2f* pst = (v2f*)&lds[POS_BASE + (e_local * ROWS + n) * 2];
    pst[0] = (v2f){p0, p1};
  }
  __syncthreads();

  // pull all 6 real rows of this element into registers (b64 broadcasts)
  v4f fr0[NBOX], fr1[NBOX];
  float msk[NBOX];
#pragma unroll
  for (int m = 0; m < NBOX; ++m) {
    const v2f* fp = (const v2f*)&lds[FEAT_BASE + (e_local * ROWS + m) * FEAT_STRIDE];
    v2f f0 = fp[0], f1 = fp[1], f2 = fp[2], f3 = fp[3];
    fr0[m] = (v4f){f0.x, f0.y, f1.x, f1.y};
    fr1[m] = (v4f){f2.x, f2.y, f3.x, f3.y};
    const v2f* qp = (const v2f*)&lds[POS_BASE + (e_local * ROWS + m) * 2];
    v2f q = qp[0];
    const float dx = p0 - q.x, dy = p1 - q.y;
    msk[m] = (dx * dx + dy * dy > 16.f) ? -__builtin_inff() : 0.f;
  }

  const v4f* M4 = (const v4f*)&lds[LDS_M];
  const v4f* U4 = (const v4f*)&lds[LDS_U];
  const v4f* V4 = (const v4f*)&lds[LDS_V];
  const v4f* G4 = (const v4f*)&lds[LDS_GCN];

  float accum[NFG];
#pragma unroll
  for (int o = 0; o < NFG; ++o) accum[o] = 0.f;

#pragma unroll
  for (int g = 0; g < NG; ++g) {
    // su = c_g + u_g . feat_n
    v4f u0 = U4[2 * g], u1 = U4[2 * g + 1];
    v4f up = u0 * (v4f){feat[0], feat[1], feat[2], feat[3]} +
             u1 * (v4f){feat[4], feat[5], feat[6], feat[7]};
    float su = lds[LDS_C + g] + up[0] + up[1] + up[2] + up[3];

    // Tv = v_g + feat_n M_g   (row-wise over M_g, 16B broadcast reads)
    v4f t0 = V4[2 * g], t1 = V4[2 * g + 1];
#pragma unroll
    for (int i = 0; i < NFG; ++i) {
      v4f m0 = M4[g * 16 + 2 * i], m1 = M4[g * 16 + 2 * i + 1];
      t0 += feat[i] * m0;
      t1 += feat[i] * m1;
    }

    // sim row + masked softmax (row-local, register-only)
    float s[NBOX];
    float mx = -__builtin_inff();
#pragma unroll
    for (int m = 0; m < NBOX; ++m) {
      v4f pr = t0 * fr0[m] + t1 * fr1[m];
      float d = su + pr[0] + pr[1] + pr[2] + pr[3];
      d = d * 0.0625f + msk[m];
      s[m] = d;
      mx = fmaxf(mx, d);
    }
    float den = 0.f;
#pragma unroll
    for (int m = 0; m < NBOX; ++m) { s[m] = __expf(s[m] - mx); den += s[m]; }
    const float inv = 1.f / den;

    // agg = relation @ feat (registers)
    v4f a0 = {0.f, 0.f, 0.f, 0.f}, a1 = {0.f, 0.f, 0.f, 0.f};
#pragma unroll
    for (int m = 0; m < NBOX; ++m) {
      const float r = s[m] * inv;
      a0 += r * fr0[m];
      a1 += r * fr1[m];
    }

    // one = relu(agg @ W_gcn^T); accumulate mean over g
#pragma unroll
    for (int o = 0; o < NFG; ++o) {
      v4f g0 = G4[g * 16 + 2 * o], g1 = G4[g * 16 + 2 * o + 1];
      v4f pr = a0 * g0 + a1 * g1;
      accum[o] += fmaxf(pr[0] + pr[1] + pr[2] + pr[3], 0.f);
    }
  }

  // states = mean_g + residual; max-pool over rows via 8-lane shuffles
  float st[NFG];
#pragma unroll
  for (int o = 0; o < NFG; ++o) {
    const float xo = (o < 4) ? x0[o] : x1[o - 4];
    st[o] = rowvalid ? (0.25f * accum[o] + xo) : -__builtin_inff();
  }
#pragma unroll
  for (int o = 0; o < NFG; ++o) {
    float v = st[o];
    v = fmaxf(v, __shfl_xor(v, 1, 8));
    v = fmaxf(v, __shfl_xor(v, 2, 8));
    v = fmaxf(v, __shfl_xor(v, 4, 8));
    st[o] = v;
  }

  if (n < NACT && e < (long)Btot) {
    const v4f* A4 = (const v4f*)&lds[LDS_ACT];
    v4f w0 = A4[2 * n], w1 = A4[2 * n + 1];
    v4f pr = w0 * (v4f){st[0], st[1], st[2], st[3]} +
             w1 * (v4f){st[4], st[5], st[6], st[7]};
    out[e * NACT + n] = lds[LDS_BACT + n] + pr[0] + pr[1] + pr[2] + pr[3];
  }
}

extern "C" void kernel_launch(void* const* d_in, const int* in_sizes, int n_in,
                              void* d_out, int out_size, void* d_ws, size_t ws_size,
                              hipStream_t stream) {
  const float* bf   = (const float*)d_in[0];   // [B,6,8]
  const float* bp   = (const float*)d_in[1];   // [B,6,2]
  const float* Wext = (const float*)d_in[2];   // [8,8]
  const float* bext = (const float*)d_in[3];   // [8]
  const float* Wt   = (const float*)d_in[4];   // [4,256,8]
  const float* bth  = (const float*)d_in[5];   // [4,256]
  const float* Wp   = (const float*)d_in[6];   // [4,256,8]
  const float* bph  = (const float*)d_in[7];   // [4,256]
  const float* Wgcn = (const float*)d_in[8];   // [4,8,8]
  const float* Wact = (const float*)d_in[9];   // [5,8]
  const float* bact = (const float*)d_in[10];  // [5]
  float* out = (float*)d_out;
  float* ws  = (float*)d_ws;

  const int Btot = in_sizes[0] / (NBOX * NFG);

  gcn_precompute_wmma<<<1, NG * 32, 0, stream>>>(Wt, bth, Wp, bph, ws);

  const int blocks = (Btot + EPB - 1) / EPB;
  gcn_main_kernel<<<blocks, 256, 0, stream>>>(bf, bp, Wext, bext, Wgcn, Wact,
                                              bact, ws, out, Btot);
}